// BandSplitLayer_42563125903608
// MI455X (gfx1250) — compile-verified
//
#include <hip/hip_runtime.h>

typedef float v2f __attribute__((ext_vector_type(2)));
typedef float v8f __attribute__((ext_vector_type(8)));

#define EPS 1e-3f
#define NBANDS 28
#define DOUT 128
#define FREQ 257
#define NROWS 16000           // B*T = 16*1000
#define NTILES 1000           // NROWS/16
#define WAVES_PER_BLOCK 4
#define GRID_Y 50             // 50*4 = 200 waves per band -> 5 tiles each

// One wave: one band, strided 16-row tiles, all 128 output cols via 8 WMMA N-tiles.
// LayerNorm is lane-local. gamma folded into B fragments; beta & bias folded into
// the accumulator init. All runtime "half" selection is done via pointer offsets
// (address math) -- never via dynamic register-array indexing.
template<int CB>
__device__ __forceinline__ void band_worker(
    const float* __restrict__ x,
    const float* __restrict__ gamma, const float* __restrict__ beta,
    const float* __restrict__ W,     const float* __restrict__ bias,
    float* __restrict__ out,
    int band, int start_bin, int local, int tile0, int tstride)
{
    constexpr int KS = CB / 4;            // K-steps of 4 per WMMA chain
    const int lane = threadIdx.x & 31;
    const int half = lane >> 4;           // 0: lanes 0-15, 1: lanes 16-31
    const int nl   = lane & 15;

    const float* __restrict__ Wb = W + local * CB * DOUT;   // [CB][128]
    const float* __restrict__ Wh = Wb + 2 * half * DOUT;    // rows shifted by lane-half

    // gammas needed by this lane-half, loaded at compile-time offsets off a
    // runtime-shifted pointer (no dynamic register indexing).
    const float* __restrict__ gp = gamma + local * CB + 2 * half;
    float ga[KS], gb[KS];
#pragma unroll
    for (int kk = 0; kk < KS; ++kk) {
        ga[kk] = gp[4 * kk];
        gb[kk] = gp[4 * kk + 1];
    }

    float bt[CB];
#pragma unroll
    for (int j = 0; j < CB; ++j) bt[j] = beta[local * CB + j];

    // B fragments: 4x16 f32 tile per (n, kk); lane-half selects rows {2h,2h+1}
    // of each K-group, mirrored with the A layout.
    v2f Bf[8][KS];
#pragma unroll
    for (int n = 0; n < 8; ++n) {
        const int col = n * 16 + nl;
#pragma unroll
        for (int kk = 0; kk < KS; ++kk) {
            Bf[n][kk].x = ga[kk] * Wh[(4 * kk    ) * DOUT + col];
            Bf[n][kk].y = gb[kk] * Wh[(4 * kk + 1) * DOUT + col];
        }
    }

    // bias' per output column: b[col] + sum_j beta_j * W[j][col]
    float bv[8];
#pragma unroll
    for (int n = 0; n < 8; ++n) {
        const int col = n * 16 + nl;
        float s = bias[local * DOUT + col];
#pragma unroll
        for (int j = 0; j < CB; ++j) s += bt[j] * Wb[j * DOUT + col];
        bv[n] = s;
    }

    const int binoff = start_bin + local * CB;

    for (int tile = tile0; tile < NTILES; tile += tstride) {
        const int s0 = tile * 16;
        // 32-bit flat input offset (max ~4.1M elements)
        const float* __restrict__ xr  = x + ((s0 + nl) * FREQ + binoff);
        const float* __restrict__ xra = xr + 2 * half;   // A-fragment view

        float xv[CB];
#pragma unroll
        for (int j = 0; j < CB; ++j) xv[j] = xr[j];

        float mean = 0.f;
#pragma unroll
        for (int j = 0; j < CB; ++j) mean += xv[j];
        mean *= (1.0f / CB);

        float var = 0.f;
#pragma unroll
        for (int j = 0; j < CB; ++j) { float d = xv[j] - mean; var += d * d; }
        var *= (1.0f / CB);

        const float inv = rsqrtf(var + EPS);

        // A fragments (16x4 f32): element (4kk + 2*half + {0,1}) via pointer,
        // compile-time offsets -> plain loads (L0 hits), zero cndmask chains.
        v2f Af[KS];
#pragma unroll
        for (int kk = 0; kk < KS; ++kk) {
            Af[kk].x = (xra[4 * kk    ] - mean) * inv;
            Af[kk].y = (xra[4 * kk + 1] - mean) * inv;
        }

        v8f acc[8];
#pragma unroll
        for (int n = 0; n < 8; ++n)
#pragma unroll
            for (int r = 0; r < 8; ++r) acc[n][r] = bv[n];

        // K-chained WMMA: 8 independent N-tiles interleaved for pipelining.
#pragma unroll
        for (int kk = 0; kk < KS; ++kk)
#pragma unroll
            for (int n = 0; n < 8; ++n)
                acc[n] = __builtin_amdgcn_wmma_f32_16x16x4_f32(
                    false, Af[kk], false, Bf[n][kk],
                    (short)0, acc[n], false, false);

        // One base address per tile; all 64 stores use immediate offsets.
        // D layout: VGPR r -> row s0 + r + 8*half, N = lane%16.
        float* __restrict__ op =
            out + (((s0 + 8 * half) * NBANDS + band) * DOUT + nl);
#pragma unroll
        for (int n = 0; n < 8; ++n)
#pragma unroll
            for (int r = 0; r < 8; ++r)
                __builtin_nontemporal_store(acc[n][r],
                    op + (n * 16 + r * NBANDS * DOUT));
    }
}

__global__ __launch_bounds__(128)
void bandsplit_wmma_kernel(
    const float* __restrict__ x,
    const float* __restrict__ g1, const float* __restrict__ be1,
    const float* __restrict__ W1, const float* __restrict__ bi1,
    const float* __restrict__ g2, const float* __restrict__ be2,
    const float* __restrict__ W2, const float* __restrict__ bi2,
    const float* __restrict__ g3, const float* __restrict__ be3,
    const float* __restrict__ W3, const float* __restrict__ bi3,
    float* __restrict__ out)
{
    const int band = blockIdx.x;                       // 0..27 (uniform per block)
    // wave id is wave-uniform: pin it to an SGPR so the tile loop is scalar
    const int wave    = __builtin_amdgcn_readfirstlane(threadIdx.x >> 5);
    const int tile0   = blockIdx.y * WAVES_PER_BLOCK + wave;
    const int tstride = gridDim.y * WAVES_PER_BLOCK;

    if (band < 8)
        band_worker<4 >(x, g1, be1, W1, bi1, out, band,   0, band,      tile0, tstride);
    else if (band < 20)
        band_worker<8 >(x, g2, be2, W2, bi2, out, band,  32, band - 8,  tile0, tstride);
    else
        band_worker<16>(x, g3, be3, W3, bi3, out, band, 128, band - 20, tile0, tstride);
}

extern "C" void kernel_launch(void* const* d_in, const int* in_sizes, int n_in,
                              void* d_out, int out_size, void* d_ws, size_t ws_size,
                              hipStream_t stream) {
    const float* x   = (const float*)d_in[0];
    const float* g1  = (const float*)d_in[1];
    const float* be1 = (const float*)d_in[2];
    const float* W1  = (const float*)d_in[3];
    const float* bi1 = (const float*)d_in[4];
    const float* g2  = (const float*)d_in[5];
    const float* be2 = (const float*)d_in[6];
    const float* W2  = (const float*)d_in[7];
    const float* bi2 = (const float*)d_in[8];
    const float* g3  = (const float*)d_in[9];
    const float* be3 = (const float*)d_in[10];
    const float* W3  = (const float*)d_in[11];
    const float* bi3 = (const float*)d_in[12];
    float* out = (float*)d_out;

    dim3 grid(NBANDS, GRID_Y);
    dim3 block(32 * WAVES_PER_BLOCK);
    bandsplit_wmma_kernel<<<grid, block, 0, stream>>>(
        x, g1, be1, W1, bi1, g2, be2, W2, bi2, g3, be3, W3, bi3, out);
}